// ShuffledDBN_17231408791710
// MI455X (gfx1250) — compile-verified
//
#include <hip/hip_runtime.h>
#include <hip/hip_bf16.h>

typedef __attribute__((ext_vector_type(2))) float v2f;
typedef __attribute__((ext_vector_type(8))) float v8f;

#define NROWS 32768
#define FDIM  2048
#define DGRP  64
#define NGRP  32
#define LSTR  68          // LDS row stride (64 + 4 pad, bank-conflict free)
#define TILE_R 64         // rows per LDS tile
#define CHUNKS 32         // blocks per group along N
#define ROWS_PER_BLOCK (NROWS / CHUNKS)   // 1024
#define NS_ITERS 18

// ---------------------------------------------------------------------------
// 64x64 fp32 matmul in LDS using V_WMMA_F32_16X16X4_F32.
// 8 waves (256 threads), each wave computes two 16x16 tiles of C = A * B.
// A,B,C are 64x64 matrices in LDS with row stride LSTR.
// ---------------------------------------------------------------------------
__device__ __forceinline__ void mm64(const float* A, const float* B, float* C,
                                     int lane, int wave) {
  const int m = lane & 15;
  const int half = lane >> 4;
#pragma unroll
  for (int t = 0; t < 2; ++t) {
    const int ti = wave * 2 + t;
    const int i0 = (ti >> 2) * 16;
    const int j0 = (ti & 3) * 16;
    v8f acc = {};
#pragma unroll
    for (int k0 = 0; k0 < DGRP; k0 += 4) {
      const int ka = k0 + half * 2;          // lane-half selects K pair
      v2f a, b;
      a.x = A[(i0 + m) * LSTR + ka];         // A[m][ka], A[m][ka+1]
      a.y = A[(i0 + m) * LSTR + ka + 1];
      b.x = B[ka * LSTR + j0 + m];           // B[ka][n], B[ka+1][n] (n == lane&15)
      b.y = B[(ka + 1) * LSTR + j0 + m];
      acc = __builtin_amdgcn_wmma_f32_16x16x4_f32(false, a, false, b,
                                                  (short)0, acc, false, false);
    }
#pragma unroll
    for (int v = 0; v < 8; ++v)
      C[(i0 + v + 8 * half) * LSTR + j0 + m] = acc[v];   // D layout: m=v+8*half, n=lane&15
  }
}

// ---------------------------------------------------------------------------
// Kernel 0: zero Gram + colsum accumulators in workspace.
// ---------------------------------------------------------------------------
__global__ void zero_stats_kernel(float* ws, int count) {
  int i = blockIdx.x * blockDim.x + threadIdx.x;
  if (i < count) ws[i] = 0.f;
}

// ---------------------------------------------------------------------------
// Kernel 1: per-group gathered Gram (X^T X) + column sums.
// grid = (NGRP, CHUNKS): blockIdx.x = group (fastest) so the 32 groups that
// share one row window are scheduled together -> every cacheline of that
// window is fully consumed while it is resident in L2.
// block = 256 (8 waves).
// ---------------------------------------------------------------------------
__global__ void gram_kernel(const float* __restrict__ x,
                            const int* __restrict__ sidx,
                            float* __restrict__ gram,
                            float* __restrict__ colsum) {
  __shared__ float tile[TILE_R][LSTR];
  __shared__ int cols[DGRP];
  const int g = blockIdx.x;
  const int t = threadIdx.x;
  if (t < DGRP) cols[t] = sidx[g * DGRP + t];
  __syncthreads();

  const int lane = t & 31, wave = t >> 5;
  const int m = lane & 15, half = lane >> 4;
  const int tiA = wave * 2, tiB = wave * 2 + 1;
  const int i0a = (tiA >> 2) * 16, j0a = (tiA & 3) * 16;
  const int i0b = (tiB >> 2) * 16, j0b = (tiB & 3) * 16;

  v8f acc0 = {}, acc1 = {};
  float csum = 0.f;
  const int c  = t & 63;   // column this thread stages
  const int r0 = t >> 6;   // 0..3
  const long rowbase = (long)blockIdx.y * ROWS_PER_BLOCK;

  for (int tb = 0; tb < ROWS_PER_BLOCK; tb += TILE_R) {
    // Gathered (shuffled) load of a 64x64 row tile into LDS, fused colsum.
#pragma unroll
    for (int r = r0; r < TILE_R; r += 4) {
      float v = x[(size_t)(rowbase + tb + r) * FDIM + cols[c]];
      tile[r][c] = v;
      csum += v;
    }
    __syncthreads();
    // Gram tiles: D += (X^T tile)(X tile); K dimension = rows of the tile.
#pragma unroll
    for (int k0 = 0; k0 < TILE_R; k0 += 4) {
      const int ka = k0 + half * 2;
      v2f a, b;
      a.x = tile[ka][i0a + m];  a.y = tile[ka + 1][i0a + m];
      b.x = tile[ka][j0a + m];  b.y = tile[ka + 1][j0a + m];
      acc0 = __builtin_amdgcn_wmma_f32_16x16x4_f32(false, a, false, b,
                                                   (short)0, acc0, false, false);
      a.x = tile[ka][i0b + m];  a.y = tile[ka + 1][i0b + m];
      b.x = tile[ka][j0b + m];  b.y = tile[ka + 1][j0b + m];
      acc1 = __builtin_amdgcn_wmma_f32_16x16x4_f32(false, a, false, b,
                                                   (short)0, acc1, false, false);
    }
    __syncthreads();
  }

  float* gg = gram + (size_t)g * DGRP * DGRP;
#pragma unroll
  for (int v = 0; v < 8; ++v) {
    atomicAdd(&gg[(i0a + v + 8 * half) * DGRP + j0a + m], acc0[v]);
    atomicAdd(&gg[(i0b + v + 8 * half) * DGRP + j0b + m], acc1[v]);
  }
  atomicAdd(&colsum[g * DGRP + c], csum);
}

// ---------------------------------------------------------------------------
// Kernel 2: cov = (Gram - N mu mu^T) / NGRP   (reference divides by G).
// grid = NGRP, block = 256.
// ---------------------------------------------------------------------------
__global__ void finalize_cov_kernel(const float* __restrict__ gram,
                                    const float* __restrict__ colsum,
                                    float* __restrict__ cov) {
  const int g = blockIdx.x;
  const int t = threadIdx.x;
  __shared__ float mu[DGRP];
  if (t < DGRP) mu[t] = colsum[g * DGRP + t] * (1.f / (float)NROWS);
  __syncthreads();
  const float invG = 1.f / (float)NGRP;
  for (int e = t; e < DGRP * DGRP; e += 256) {
    const int i = e >> 6, j = e & 63;
    float v = gram[(size_t)g * DGRP * DGRP + e] - (float)NROWS * mu[i] * mu[j];
    cov[(size_t)g * DGRP * DGRP + e] = v * invG;
  }
}

// ---------------------------------------------------------------------------
// Kernel 3: W = cov^{-1/2} via coupled Newton-Schulz, all in LDS with WMMA.
//   Y0 = A/tr(A), Z0 = I; T = 1.5 I - 0.5 Z Y; Y <- Y T; Z <- T Z.
//   A^{-1/2} = Z / sqrt(tr(A)).
// grid = NGRP, block = 256 (8 waves). 4 rotating 64x68 LDS buffers (~68 KB).
// ---------------------------------------------------------------------------
__global__ void ns_invsqrt_kernel(const float* __restrict__ cov,
                                  float* __restrict__ Wm) {
  __shared__ float bufs[4][DGRP * LSTR];
  __shared__ float tr_sh;
  const int g = blockIdx.x;
  const int t = threadIdx.x;
  const int lane = t & 31, wave = t >> 5;

  float* Y = bufs[0];
  float* Z = bufs[1];
  float* T = bufs[2];
  float* S = bufs[3];

  const float* Ag = cov + (size_t)g * DGRP * DGRP;
  for (int e = t; e < DGRP * DGRP; e += 256)
    Y[(e >> 6) * LSTR + (e & 63)] = Ag[e];
  __syncthreads();
  if (t == 0) {
    float tr = 0.f;
    for (int i = 0; i < DGRP; ++i) tr += Y[i * LSTR + i];
    tr_sh = tr;
  }
  __syncthreads();
  const float trace = tr_sh;
  const float inv_tr = 1.f / trace;
  for (int e = t; e < DGRP * DGRP; e += 256) {
    const int i = e >> 6, j = e & 63;
    Y[i * LSTR + j] *= inv_tr;
    Z[i * LSTR + j] = (i == j) ? 1.f : 0.f;
  }
  __syncthreads();

  for (int it = 0; it < NS_ITERS; ++it) {
    mm64(Z, Y, T, lane, wave);          // T = Z*Y
    __syncthreads();
    for (int e = t; e < DGRP * DGRP; e += 256) {
      const int i = e >> 6, j = e & 63;
      T[i * LSTR + j] = ((i == j) ? 1.5f : 0.f) - 0.5f * T[i * LSTR + j];
    }
    __syncthreads();
    mm64(Y, T, S, lane, wave);          // newY = Y*T
    __syncthreads();
    mm64(T, Z, Y, lane, wave);          // newZ = T*Z  (Y's data no longer needed)
    __syncthreads();
    float* newY = S;
    float* newZ = Y;
    S = Z;                              // old Z becomes scratch
    Y = newY;
    Z = newZ;                           // T buffer reused next iter
  }

  const float sc = rsqrtf(trace);       // A^{-1/2} = Z / sqrt(tr)
  for (int e = t; e < DGRP * DGRP; e += 256)
    Wm[(size_t)g * DGRP * DGRP + e] = Z[(e >> 6) * LSTR + (e & 63)] * sc;
}

// ---------------------------------------------------------------------------
// Kernel 4: Y = (X_gathered - mu) * W, scattered back to original columns.
// grid = (NGRP, CHUNKS): groups of the same row window scheduled together
// (L2 reuse for both the gathered reads and the scattered writes).
// block = 256 (8 waves).
// ---------------------------------------------------------------------------
__global__ void apply_kernel(const float* __restrict__ x,
                             const int* __restrict__ sidx,
                             const float* __restrict__ Wm,
                             const float* __restrict__ colsum,
                             float* __restrict__ out) {
  __shared__ float tile[TILE_R][LSTR];
  __shared__ float Wls[DGRP][LSTR];
  __shared__ float mu[DGRP];
  __shared__ int cols[DGRP];
  const int g = blockIdx.x;
  const int t = threadIdx.x;
  if (t < DGRP) {
    cols[t] = sidx[g * DGRP + t];
    mu[t] = colsum[g * DGRP + t] * (1.f / (float)NROWS);
  }
  for (int e = t; e < DGRP * DGRP; e += 256)
    Wls[e >> 6][e & 63] = Wm[(size_t)g * DGRP * DGRP + e];
  __syncthreads();

  const int lane = t & 31, wave = t >> 5;
  const int m = lane & 15, half = lane >> 4;
  const int c  = t & 63;
  const int rr = t >> 6;
  const long rowbase = (long)blockIdx.y * ROWS_PER_BLOCK;

  for (int tb = 0; tb < ROWS_PER_BLOCK; tb += TILE_R) {
    // gathered + centered tile
#pragma unroll
    for (int r = rr; r < TILE_R; r += 4)
      tile[r][c] = x[(size_t)(rowbase + tb + r) * FDIM + cols[c]] - mu[c];
    __syncthreads();
    // each wave: two 16x16 output tiles of  tile(64x64) @ W(64x64)
#pragma unroll
    for (int tt = 0; tt < 2; ++tt) {
      const int ti = wave * 2 + tt;
      const int r0 = (ti >> 2) * 16;    // row offset within tile
      const int j0 = (ti & 3) * 16;     // output-feature offset
      v8f acc = {};
#pragma unroll
      for (int k0 = 0; k0 < DGRP; k0 += 4) {
        const int ka = k0 + half * 2;
        v2f a, b;
        a.x = tile[r0 + m][ka];         // A[m][k] = Xc[row r0+m][feat k]
        a.y = tile[r0 + m][ka + 1];
        b.x = Wls[ka][j0 + m];          // B[k][n] = W[k][j0+n]
        b.y = Wls[ka + 1][j0 + m];
        acc = __builtin_amdgcn_wmma_f32_16x16x4_f32(false, a, false, b,
                                                    (short)0, acc, false, false);
      }
      // scatter: out[row, shuffle_idx[g*64 + j]] = y
      const int ocol = cols[j0 + m];
#pragma unroll
      for (int v = 0; v < 8; ++v) {
        const long row = rowbase + tb + r0 + v + 8 * half;
        out[(size_t)row * FDIM + ocol] = acc[v];
      }
    }
    __syncthreads();
  }
}

// ---------------------------------------------------------------------------
// Host-side launch. Workspace layout (floats):
//   [0)              gram    : 32*64*64 = 131072
//   [131072)         colsum  : 2048
//   [133120)         cov     : 131072
//   [264192)         W       : 131072
// total = 395264 floats ≈ 1.6 MB
// ---------------------------------------------------------------------------
extern "C" void kernel_launch(void* const* d_in, const int* in_sizes, int n_in,
                              void* d_out, int out_size, void* d_ws, size_t ws_size,
                              hipStream_t stream) {
  const float* x   = (const float*)d_in[0];
  const int* sidx  = (const int*)d_in[1];
  float* out       = (float*)d_out;

  float* ws     = (float*)d_ws;
  float* gram   = ws;
  float* colsum = ws + 131072;
  float* cov    = ws + 133120;
  float* Wm     = ws + 264192;

  const int zcount = 133120;
  zero_stats_kernel<<<(zcount + 255) / 256, 256, 0, stream>>>(ws, zcount);

  gram_kernel<<<dim3(NGRP, CHUNKS), 256, 0, stream>>>(x, sidx, gram, colsum);

  finalize_cov_kernel<<<NGRP, 256, 0, stream>>>(gram, colsum, cov);

  ns_invsqrt_kernel<<<NGRP, 256, 0, stream>>>(cov, Wm);

  apply_kernel<<<dim3(NGRP, CHUNKS), 256, 0, stream>>>(x, sidx, Wm, colsum, out);
}